// FullAdaptiveModel_76493367541903
// MI455X (gfx1250) — compile-verified
//
#include <hip/hip_runtime.h>
#include <cmath>

#define B_     128
#define L_     2048
#define HID_   64
#define HID2_  128
#define HALF_  32
#define VOCAB_ 64

typedef float v2f __attribute__((ext_vector_type(2)));
typedef float v8f __attribute__((ext_vector_type(8)));

__device__ __forceinline__ float wsum32(float v) {
  v += __shfl_xor(v, 16, 32);
  v += __shfl_xor(v, 8, 32);
  v += __shfl_xor(v, 4, 32);
  v += __shfl_xor(v, 2, 32);
  v += __shfl_xor(v, 1, 32);
  return v;
}

// ---------------------------------------------------------------------------
// Phase 1: token-parallel  embed -> FFN(+relu) -> residual -> LN -> sem/epi
// One wave32 per 16-token tile. All GEMMs via V_WMMA_F32_16X16X4_F32 (fp32).
// ---------------------------------------------------------------------------
__global__ __launch_bounds__(32) void token_kernel(
    const int* __restrict__ seq, const float* __restrict__ embed_w,
    const float* __restrict__ w1, const float* __restrict__ b1,
    const float* __restrict__ w2, const float* __restrict__ b2,
    const float* __restrict__ lnw, const float* __restrict__ lnb,
    const float* __restrict__ semw, const float* __restrict__ epiw,
    float* __restrict__ ks, float* __restrict__ ke)
{
  __shared__ float hbuf[16][65];    // 16 tokens x 64 (pad 65: conflict-free)
  __shared__ float c1buf[16][132];  // 16 x 128 (pad 132: stride 4 banks)

  const int lane = threadIdx.x;
  const int tile = blockIdx.x;
  const int b  = tile >> 7;          // L_/16 == 128 tiles per batch row
  const int l0 = (tile & 127) << 4;

  // CDNA5 prefetch (global_prefetch_b8): warm weights into near caches
  __builtin_prefetch(w1 + lane * 64, 0, 1);
  __builtin_prefetch(w2 + lane * 128, 0, 1);

  // --- gather embeddings: lane -> row lane>>1, half (lane&1)*32
  {
    const int r  = lane >> 1;
    const int c0 = (lane & 1) * 32;
    const int tok = seq[b * L_ + l0 + r];
    const float* erow = embed_w + tok * HID_ + c0;
    #pragma unroll
    for (int c = 0; c < 32; ++c) hbuf[r][c0 + c] = erow[c];
  }
  __syncthreads();

  // WMMA f32 16x16x4 fragment indexing (per ISA layout tables):
  //   A: lanes 0-15 hold M=lane, K={k0,k0+1}; lanes 16-31 hold M=lane-16, K={k0+2,k0+3}
  //   B mirrors A in K; C/D: VGPR r -> M=r (lanes<16) / r+8 (lanes>=16), N=lane%16
  const int aM = lane & 15;
  const int kb = (lane >> 4) << 1;        // 0 or 2
  const int mbase = (lane < 16) ? 0 : 8;

  // ---- GEMM1: c1 = relu(h @ w1^T + b1)   (16x64)x(64x128)
  #pragma unroll
  for (int nt = 0; nt < 8; ++nt) {
    v8f acc = {0.f,0.f,0.f,0.f,0.f,0.f,0.f,0.f};
    const int n = nt * 16 + (lane & 15);
    const float* wrow = w1 + n * HID_;   // B[k][n] = w1[n*HID + k]
    #pragma unroll
    for (int k0 = 0; k0 < HID_; k0 += 4) {
      v2f a, bb;
      a.x  = hbuf[aM][k0 + kb];
      a.y  = hbuf[aM][k0 + kb + 1];
      bb.x = wrow[k0 + kb];
      bb.y = wrow[k0 + kb + 1];
      acc = __builtin_amdgcn_wmma_f32_16x16x4_f32(false, a, false, bb,
                                                  (short)0, acc, false, false);
    }
    const float bias = b1[n];
    #pragma unroll
    for (int rr = 0; rr < 8; ++rr)
      c1buf[mbase + rr][n] = fmaxf(acc[rr] + bias, 0.f);
  }
  __syncthreads();

  // ---- GEMM2: x = c1 @ w2^T + b2 + h   (16x128)x(128x64), residual in-place
  #pragma unroll
  for (int nt = 0; nt < 4; ++nt) {
    v8f acc = {0.f,0.f,0.f,0.f,0.f,0.f,0.f,0.f};
    const int n = nt * 16 + (lane & 15);
    const float* wrow = w2 + n * HID2_;
    #pragma unroll
    for (int k0 = 0; k0 < HID2_; k0 += 4) {
      v2f a, bb;
      a.x  = c1buf[aM][k0 + kb];
      a.y  = c1buf[aM][k0 + kb + 1];
      bb.x = wrow[k0 + kb];
      bb.y = wrow[k0 + kb + 1];
      acc = __builtin_amdgcn_wmma_f32_16x16x4_f32(false, a, false, bb,
                                                  (short)0, acc, false, false);
    }
    const float bias = b2[n];
    #pragma unroll
    for (int rr = 0; rr < 8; ++rr) {
      const int m = mbase + rr;
      hbuf[m][n] = acc[rr] + bias + hbuf[m][n];   // each (m,n) touched by one lane
    }
  }
  __syncthreads();

  // ---- LayerNorm over HID=64 per token (lanes 0-15 each own one row)
  if (lane < 16) {
    float mu = 0.f;
    #pragma unroll
    for (int c = 0; c < HID_; ++c) mu += hbuf[lane][c];
    mu *= (1.0f / HID_);
    float var = 0.f;
    #pragma unroll
    for (int c = 0; c < HID_; ++c) { float d = hbuf[lane][c] - mu; var += d * d; }
    var *= (1.0f / HID_);
    const float rstd = 1.0f / sqrtf(var + 1e-5f);
    #pragma unroll
    for (int c = 0; c < HID_; ++c)
      hbuf[lane][c] = (hbuf[lane][c] - mu) * rstd * lnw[c] + lnb[c];
  }
  __syncthreads();

  // ---- GEMM3: [ks|ke] = h_ln @ [sem_w^T | epi_w^T]   (16x64)x(64x64)
  #pragma unroll
  for (int nt = 0; nt < 4; ++nt) {
    v8f acc = {0.f,0.f,0.f,0.f,0.f,0.f,0.f,0.f};
    const int n = nt * 16 + (lane & 15);
    const float* P = (n < HALF_) ? semw : epiw;
    const int j = n & 31;
    const float* wrow = P + j * HID_;
    #pragma unroll
    for (int k0 = 0; k0 < HID_; k0 += 4) {
      v2f a, bb;
      a.x  = hbuf[aM][k0 + kb];
      a.y  = hbuf[aM][k0 + kb + 1];
      bb.x = wrow[k0 + kb];
      bb.y = wrow[k0 + kb + 1];
      acc = __builtin_amdgcn_wmma_f32_16x16x4_f32(false, a, false, bb,
                                                  (short)0, acc, false, false);
    }
    float* dst = (n < HALF_) ? ks : ke;
    #pragma unroll
    for (int rr = 0; rr < 8; ++rr) {
      const int l = l0 + mbase + rr;
      dst[((size_t)b * L_ + l) * HALF_ + j] = acc[rr];
    }
  }
}

// ---------------------------------------------------------------------------
// Phase 2: sequential fast-weight scan. One wave32 per batch; lane i owns
// row i of M_s and M_e in registers (32+32 VGPRs of state). Then the head.
// ---------------------------------------------------------------------------
__global__ __launch_bounds__(32) void scan_kernel(
    const float* __restrict__ ks, const float* __restrict__ ke,
    const float* __restrict__ rpw, const float* __restrict__ rpb,
    const float* __restrict__ outw, const float* __restrict__ outb,
    float* __restrict__ out, float c_const)
{
  __shared__ float s_kns[32];
  __shared__ float s_kne[32];
  __shared__ float s_feat[64];
  __shared__ float s_r[64];

  const int lane = threadIdx.x;
  const int b = blockIdx.x;
  const float* ksb = ks + (size_t)b * L_ * HALF_;
  const float* keb = ke + (size_t)b * L_ * HALF_;

  float Ms[32], Me[32];
  #pragma unroll
  for (int j = 0; j < 32; ++j) { Ms[j] = 0.f; Me[j] = 0.f; }

  const float inv_norm_floor = 1e-12f;
  const float th2 = 0.4f * 0.4f;   // compare squared norms

  for (int t = 0; t < L_ - 1; ++t) {
    const float ksv = ksb[t * HALF_ + lane];   // lane = component index
    const float kev = keb[t * HALF_ + lane];
    const float ks2 = wsum32(ksv * ksv);
    const float ke2 = wsum32(kev * kev);
    const float kns = ksv / fmaxf(sqrtf(ks2), inv_norm_floor);
    const float kne = kev / fmaxf(sqrtf(ke2), inv_norm_floor);
    s_kns[lane] = kns;
    s_kne[lane] = kne;
    __syncthreads();

    // rs[i] = ks[i] - sum_j Ms[i][j]*kns[j]  (lane i holds row i)
    float accs = 0.f, acce = 0.f;
    #pragma unroll
    for (int j = 0; j < 32; ++j) {
      accs = fmaf(Ms[j], s_kns[j], accs);
      acce = fmaf(Me[j], s_kne[j], acce);
    }
    const float rs = ksv - accs;
    const float re = kev - acce;
    const float rs2 = wsum32(rs * rs);
    const float re2 = wsum32(re * re);

    // wave-uniform gate (reduced scalars are identical on all lanes)
    if ((rs2 >= th2 * ks2) || (re2 >= th2 * ke2)) {
      const float wt = (float)(t + 1) / (float)L_;
      const float cs = c_const * rs;
      const float ce = c_const * wt * re;
      #pragma unroll
      for (int j = 0; j < 32; ++j) {
        Ms[j] = fmaf(cs, s_kns[j], Ms[j]);
        Me[j] = fmaf(ce, s_kne[j], Me[j]);
      }
    }
    __syncthreads();   // protect s_kns/s_kne before next iteration's write
  }

  // ---- head: q projections are ks/ke at l = L-1 (same linear maps)
  s_kns[lane] = ksb[(size_t)(L_ - 1) * HALF_ + lane];
  s_kne[lane] = keb[(size_t)(L_ - 1) * HALF_ + lane];
  __syncthreads();

  float cs = 0.f, ce = 0.f;
  #pragma unroll
  for (int j = 0; j < 32; ++j) {
    cs = fmaf(Ms[j], s_kns[j], cs);
    ce = fmaf(Me[j], s_kne[j], ce);
  }
  s_feat[lane] = cs;
  s_feat[32 + lane] = ce;
  __syncthreads();

  // r = feats @ rp_w^T + rp_b   (64 outputs, 2 per lane)
  #pragma unroll
  for (int mi = 0; mi < 2; ++mi) {
    const int m = lane + mi * 32;
    float acc = rpb[m];
    const float* wrow = rpw + m * 64;
    #pragma unroll
    for (int k = 0; k < 64; ++k) acc = fmaf(s_feat[k], wrow[k], acc);
    s_r[m] = acc;
  }
  __syncthreads();

  // logits = r @ out_w^T + out_b
  #pragma unroll
  for (int vi = 0; vi < 2; ++vi) {
    const int v = lane + vi * 32;
    float acc = outb[v];
    const float* wrow = outw + v * 64;
    #pragma unroll
    for (int k = 0; k < 64; ++k) acc = fmaf(s_r[k], wrow[k], acc);
    out[b * VOCAB_ + v] = acc;
  }
}

extern "C" void kernel_launch(void* const* d_in, const int* in_sizes, int n_in,
                              void* d_out, int out_size, void* d_ws, size_t ws_size,
                              hipStream_t stream) {
  (void)in_sizes; (void)n_in; (void)out_size; (void)ws_size;
  const int*   seq     = (const int*)d_in[0];
  const float* embed_w = (const float*)d_in[1];
  const float* ff_w1   = (const float*)d_in[2];
  const float* ff_b1   = (const float*)d_in[3];
  const float* ff_w2   = (const float*)d_in[4];
  const float* ff_b2   = (const float*)d_in[5];
  const float* ln_w    = (const float*)d_in[6];
  const float* ln_b    = (const float*)d_in[7];
  const float* sem_w   = (const float*)d_in[8];
  const float* epi_w   = (const float*)d_in[9];
  const float* rp_w    = (const float*)d_in[10];
  const float* rp_b    = (const float*)d_in[11];
  const float* out_w   = (const float*)d_in[12];
  const float* out_b   = (const float*)d_in[13];

  float* ks = (float*)d_ws;                              // B*L*32 f32 = 32 MB
  float* ke = ks + (size_t)B_ * L_ * HALF_;              // another 32 MB

  // c = 1 - 0.95^(96/L), computed in double exactly like the reference
  const float c_const = (float)(1.0 - std::pow(0.95, 96.0 / (double)L_));

  token_kernel<<<B_ * (L_ / 16), 32, 0, stream>>>(
      seq, embed_w, ff_w1, ff_b1, ff_w2, ff_b2, ln_w, ln_b, sem_w, epi_w, ks, ke);

  scan_kernel<<<B_, 32, 0, stream>>>(
      ks, ke, rp_w, rp_b, out_w, out_b, (float*)d_out, c_const);
}